// HypergraphEncoder_77077483094359
// MI455X (gfx1250) — compile-verified
//
#include <hip/hip_runtime.h>

typedef __attribute__((ext_vector_type(2))) float v2f;
typedef __attribute__((ext_vector_type(8))) float v8f;

#define DIM 128

// ---------------------------------------------------------------------------
// Stage 1: per-edge mean of member node embeddings.
// One 128-thread block per edge; member ids are block-uniform -> s_load,
// per-k row reads are 512B contiguous across the block.
// ---------------------------------------------------------------------------
__global__ __launch_bounds__(DIM) void edge_mean_k(
    const float* __restrict__ node_emb, const int* __restrict__ node_ids,
    float* __restrict__ mean, int E, int DEG)
{
    const int e = blockIdx.x;
    const int d = threadIdx.x;
    const int* ids = node_ids + (size_t)e * DEG;
    float s = 0.0f;
    for (int k = 0; k < DEG; ++k)
        s += node_emb[(size_t)ids[k] * DIM + d];
    mean[(size_t)e * DIM + d] = s / (float)DEG;
}

// ---------------------------------------------------------------------------
// Stage 2: edge_ctx = mean @ W_e^T + b_e ; edge_out = edge_emb + edge_ctx
// WMMA f32 16x16x4. 8 waves per block, wave w computes cols [16w,16w+16).
// A 16x4 layout: lanes 0-15 -> rows, VGPR0={K0,K2}, VGPR1={K1,K3} by half.
// B 4x16 = W^T -> per-lane load pattern identical to A on rows of W.
// ---------------------------------------------------------------------------
__global__ __launch_bounds__(256) void edge_gemm_k(
    const float* __restrict__ A, const float* __restrict__ edge_emb,
    const float* __restrict__ W, const float* __restrict__ b,
    float* __restrict__ edge_ctx, float* __restrict__ edge_out, int E)
{
    const int lane = threadIdx.x & 31;
    const int wv   = threadIdx.x >> 5;   // 0..7 -> col tile
    const int half = lane >> 4;
    const int c16  = lane & 15;
    const int r0 = blockIdx.x * 16;
    const int j0 = wv * 16;

    int arow = r0 + c16; if (arow >= E) arow = E - 1;   // clamp (EXEC must stay all-1)
    const float* Ap = A + (size_t)arow * DIM;
    const float* Wp = W + (size_t)(j0 + c16) * DIM;

    v8f acc = {};
#pragma unroll
    for (int k0 = 0; k0 < DIM; k0 += 4) {
        const int k = k0 + 2 * half;
        v2f av = *(const v2f*)(Ap + k);
        v2f bv = *(const v2f*)(Wp + k);
        acc = __builtin_amdgcn_wmma_f32_16x16x4_f32(false, av, false, bv,
                                                    (short)0, acc, false, false);
    }

    const int oc = j0 + c16;
    const float bias = b[oc];
#pragma unroll
    for (int v = 0; v < 8; ++v) {
        const int row = r0 + v + 8 * half;   // C/D layout: VGPR v -> M=v (+8 for hi half)
        if (row < E) {
            const size_t idx = (size_t)row * DIM + oc;
            const float val = acc[v] + bias;
            edge_ctx[idx] = val;
            edge_out[idx] = edge_emb[idx] + val;
        }
    }
}

// ---------------------------------------------------------------------------
// Stage 3: scatter-add edge_ctx rows to member nodes + degree counts.
// Edge-major: ctx row read once into registers, 32x atomic fan-out.
// ---------------------------------------------------------------------------
__global__ __launch_bounds__(DIM) void scatter_k(
    const float* __restrict__ edge_ctx, const int* __restrict__ node_ids,
    float* __restrict__ node_sum, float* __restrict__ deg, int E, int DEG)
{
    const int e = blockIdx.x;
    const int d = threadIdx.x;
    const float v = edge_ctx[(size_t)e * DIM + d];
    const int* ids = node_ids + (size_t)e * DEG;
    for (int k = 0; k < DEG; ++k)
        atomicAdd(node_sum + (size_t)ids[k] * DIM + d, v);
    if (d < DEG)
        atomicAdd(deg + ids[d], 1.0f);
}

// ---------------------------------------------------------------------------
// Stage 4: node_out = node_emb + (node_sum/(1+deg)) @ W_v^T + b_v
// Same WMMA scheme; per-row 1/(1+deg) applied to A fragments (diag-scale
// commutes with right-multiplication by W^T).
// ---------------------------------------------------------------------------
__global__ __launch_bounds__(256) void node_gemm_k(
    const float* __restrict__ S, const float* __restrict__ deg,
    const float* __restrict__ node_emb, const float* __restrict__ W,
    const float* __restrict__ b, float* __restrict__ node_out, int N)
{
    const int lane = threadIdx.x & 31;
    const int wv   = threadIdx.x >> 5;
    const int half = lane >> 4;
    const int c16  = lane & 15;
    const int r0 = blockIdx.x * 16;
    const int j0 = wv * 16;

    int arow = r0 + c16; if (arow >= N) arow = N - 1;
    const float inv = 1.0f / (1.0f + deg[arow]);
    const float* Sp = S + (size_t)arow * DIM;
    const float* Wp = W + (size_t)(j0 + c16) * DIM;

    v8f acc = {};
#pragma unroll
    for (int k0 = 0; k0 < DIM; k0 += 4) {
        const int k = k0 + 2 * half;
        v2f av = *(const v2f*)(Sp + k);
        av.x *= inv; av.y *= inv;
        v2f bv = *(const v2f*)(Wp + k);
        acc = __builtin_amdgcn_wmma_f32_16x16x4_f32(false, av, false, bv,
                                                    (short)0, acc, false, false);
    }

    const int oc = j0 + c16;
    const float bias = b[oc];
#pragma unroll
    for (int v = 0; v < 8; ++v) {
        const int row = r0 + v + 8 * half;
        if (row < N) {
            const size_t idx = (size_t)row * DIM + oc;
            node_out[idx] = node_emb[idx] + (acc[v] + bias);
        }
    }
}

// ---------------------------------------------------------------------------
// Host launch
// ---------------------------------------------------------------------------
extern "C" void kernel_launch(void* const* d_in, const int* in_sizes, int n_in,
                              void* d_out, int out_size, void* d_ws, size_t ws_size,
                              hipStream_t stream)
{
    const float* node_emb = (const float*)d_in[0];
    const float* edge_emb = (const float*)d_in[1];
    const float* W_e      = (const float*)d_in[2];
    const float* b_e      = (const float*)d_in[3];
    const float* W_v      = (const float*)d_in[4];
    const float* b_v      = (const float*)d_in[5];
    const int*   node_ids = (const int*)d_in[6];
    // d_in[7] (edge_ids) is CSR repeat(arange(E), DEG): edge e owns
    // incidences [e*DEG, (e+1)*DEG) of node_ids -- used implicitly.

    const int N   = in_sizes[0] / DIM;
    const int E   = in_sizes[1] / DIM;
    const int I   = in_sizes[6];
    const int DEG = I / E;

    float* node_out = (float*)d_out;
    float* edge_out = (float*)d_out + (size_t)N * DIM;

    // workspace layout: node_sum | deg | edge_mean | edge_ctx  (~72 MB)
    char* ws = (char*)d_ws;
    float* node_sum = (float*)ws;  ws += (size_t)N * DIM * sizeof(float);
    float* deg      = (float*)ws;  ws += (size_t)N * sizeof(float);
    float* mean     = (float*)ws;  ws += (size_t)E * DIM * sizeof(float);
    float* ectx     = (float*)ws;

    // node_sum and deg are contiguous: one memset node (graph-capturable)
    hipMemsetAsync(node_sum, 0, (size_t)N * (DIM + 1) * sizeof(float), stream);

    edge_mean_k<<<E, DIM, 0, stream>>>(node_emb, node_ids, mean, E, DEG);
    edge_gemm_k<<<(E + 15) / 16, 256, 0, stream>>>(mean, edge_emb, W_e, b_e,
                                                   ectx, edge_out, E);
    scatter_k<<<E, DIM, 0, stream>>>(ectx, node_ids, node_sum, deg, E, DEG);
    node_gemm_k<<<(N + 15) / 16, 256, 0, stream>>>(node_sum, deg, node_emb,
                                                   W_v, b_v, node_out, N);
}